// SelfAttention_75952201663253
// MI455X (gfx1250) — compile-verified
//
#include <hip/hip_runtime.h>
#include <hip/hip_bf16.h>

// Problem constants (from reference): B=8, T=1024, C=1024, H=16, D=64
#define BB 8
#define TT 1024
#define CC 1024
#define HH 16
#define DD 64

typedef __attribute__((ext_vector_type(2))) float v2f;
typedef __attribute__((ext_vector_type(8))) float v8f;

static __device__ __forceinline__ v8f wmma_f32_4(v2f a, v2f b, v8f c) {
  // D = A(16x4 f32) * B(4x16 f32) + C(16x16 f32)
  return __builtin_amdgcn_wmma_f32_16x16x4_f32(
      /*neg_a=*/false, a, /*neg_b=*/false, b,
      /*c_mod=*/(short)0, c, /*reuse_a=*/false, /*reuse_b=*/false);
}

// Low 32 bits of a flat pointer to __shared__ == LDS byte offset (ISA 10.2:
// the LDS aperture's low 32 address bits are the LDS address; async-to-LDS adds LDS_BASE in HW).
static __device__ __forceinline__ unsigned lds_off(const void* p) {
  return (unsigned)(size_t)p;
}

// CDNA5 async copy: 16 bytes per lane, global -> LDS, tracked by ASYNCcnt.
static __device__ __forceinline__ void async_copy16(unsigned lds_addr, const void* gsrc) {
  asm volatile("global_load_async_to_lds_b128 %0, %1, off"
               :: "v"(lds_addr), "v"(gsrc) : "memory");
}
static __device__ __forceinline__ void wait_async0() {
  asm volatile("s_wait_asynccnt 0x0" ::: "memory");
}

// ---------------------------------------------------------------------------
// GEMM + bias:  Cm[M,N] = A[M,K] @ Bm[K,N] + bias[N]
// 256 threads = 8 waves. Block tile 128(M) x 64(N); K staged by 16 with
// double-buffered async global->LDS copies (1 barrier per stage).
// ---------------------------------------------------------------------------
__global__ __launch_bounds__(256) void gemm_bias_kernel(
    const float* __restrict__ A, const float* __restrict__ Bm,
    const float* __restrict__ bias, float* __restrict__ Cm,
    int M, int N, int K)
{
  constexpr int TM = 128, TN = 64, TK = 16;
  constexpr int AS = TK + 4;   // 20 floats -> 80B row stride (16B aligned)
  constexpr int BS = TN + 4;   // 68 floats -> 272B row stride (16B aligned)
  __shared__ __align__(16) float Asb[2][TM * AS];
  __shared__ __align__(16) float Bsb[2][TK * BS];

  const int tid  = threadIdx.x;
  const int lane = tid & 31;
  const int wave = tid >> 5;            // 0..7
  const int half = (lane >> 4) & 1;     // wave32: K-pair / M-half selector
  const int lm   = lane & 15;

  const int tileM = blockIdx.y * TM;
  const int tileN = blockIdx.x * TN;

  // Per-thread staging coordinates (A: two 16B chunks, B: one 16B chunk)
  const int arow0 = tid >> 2, ac4 = (tid & 3) * 4;          // rows 0..63 (+64)
  const int brow  = tid >> 4, bc4 = (tid & 15) * 4;

  const v8f vzero = {0.f,0.f,0.f,0.f,0.f,0.f,0.f,0.f};
  v8f acc[4];
#pragma unroll
  for (int j = 0; j < 4; ++j) acc[j] = vzero;

  const int nstages = K / TK;

  // Prologue: issue stage 0 into buffer 0
  {
    float* Ab = Asb[0];
    float* Bb = Bsb[0];
#pragma unroll
    for (int i = 0; i < 2; ++i) {
      const int row = arow0 + 64 * i;
      async_copy16(lds_off(&Ab[row * AS + ac4]),
                   A + (size_t)(tileM + row) * K + ac4);
    }
    async_copy16(lds_off(&Bb[brow * BS + bc4]),
                 Bm + (size_t)brow * N + tileN + bc4);
  }

  for (int s = 0; s < nstages; ++s) {
    wait_async0();        // own stage-s copies landed in LDS
    __syncthreads();      // everyone's copies visible; prior readers done

    if (s + 1 < nstages) {      // overlap: issue stage s+1 while computing s
      const int k0 = (s + 1) * TK;
      float* Ab = Asb[(s + 1) & 1];
      float* Bb = Bsb[(s + 1) & 1];
#pragma unroll
      for (int i = 0; i < 2; ++i) {
        const int row = arow0 + 64 * i;
        async_copy16(lds_off(&Ab[row * AS + ac4]),
                     A + (size_t)(tileM + row) * K + k0 + ac4);
      }
      async_copy16(lds_off(&Bb[brow * BS + bc4]),
                   Bm + (size_t)(k0 + brow) * N + tileN + bc4);
    }

    const float* Ac = Asb[s & 1];
    const float* Bc = Bsb[s & 1];
    const float* arow = &Ac[(wave * 16 + lm) * AS];
#pragma unroll
    for (int kk = 0; kk < TK; kk += 4) {
      const int kr = kk + 2 * half;   // lanes 0-15: K=kk,kk+1 ; 16-31: kk+2,kk+3
      v2f a;
      a.x = arow[kr];
      a.y = arow[kr + 1];
#pragma unroll
      for (int j = 0; j < 4; ++j) {
        v2f b;
        b.x = Bc[kr       * BS + j * 16 + lm];
        b.y = Bc[(kr + 1) * BS + j * 16 + lm];
        acc[j] = wmma_f32_4(a, b, acc[j]);
      }
    }
  }

  // Epilogue: VGPR r holds row (8*half + r), col = 16j + lane%16
#pragma unroll
  for (int j = 0; j < 4; ++j) {
    const int col = tileN + j * 16 + lm;
    const float bv = bias[col];
#pragma unroll
    for (int r = 0; r < 8; ++r) {
      const int row = tileM + wave * 16 + half * 8 + r;
      Cm[(size_t)row * N + col] = acc[j][r] + bv;
    }
  }
}

// ---------------------------------------------------------------------------
// Flash attention over qkv[B,T,3C] (heads packed), writes y[B,T,C].
// 128 threads = 4 waves per (b, h, 64 q-rows). Wave owns 16 q-rows.
// K/V chunks staged via async global->LDS copies.
// ---------------------------------------------------------------------------
__global__ __launch_bounds__(128) void attn_kernel(
    const float* __restrict__ qkv, const unsigned char* __restrict__ tmask,
    float* __restrict__ y)
{
  constexpr int KC = 64;          // keys per chunk
  constexpr int SP = DD + 4;      // 68-float padded LDS row stride
  __shared__ __align__(16) float Ks[KC * SP];
  __shared__ __align__(16) float Vs[KC * SP];
  __shared__ __align__(16) float Ps[4 * 16 * SP];   // per-wave P staging
  __shared__ float keepk[KC];

  const int tid  = threadIdx.x;
  const int lane = tid & 31;
  const int wave = tid >> 5;            // 0..3
  const int half = (lane >> 4) & 1;
  const int lm   = lane & 15;

  const int qt = blockIdx.x;            // q tile of 64 rows
  const int h  = blockIdx.y;
  const int b  = blockIdx.z;

  const size_t rstride = (size_t)3 * CC;
  const int qrowA = qt * 64 + wave * 16 + lm;   // A-fragment row for this lane
  const float* qg = qkv + ((size_t)b * TT + qrowA) * rstride + (size_t)h * DD;

  // Preload Q fragments, pre-scaled by 1/sqrt(D) = 0.125
  v2f qa[16];
#pragma unroll
  for (int s = 0; s < 16; ++s) {
    const int d0 = 4 * s + 2 * half;
    qa[s].x = qg[d0]     * 0.125f;
    qa[s].y = qg[d0 + 1] * 0.125f;
  }

  const v8f vzero = {0.f,0.f,0.f,0.f,0.f,0.f,0.f,0.f};
  v8f o[4];
#pragma unroll
  for (int j = 0; j < 4; ++j) o[j] = vzero;
  float mrow[8], lrow[8];
#pragma unroll
  for (int r = 0; r < 8; ++r) { mrow[r] = -1e30f; lrow[r] = 0.f; }

  for (int kc = 0; kc < TT; kc += KC) {
    __syncthreads();     // previous chunk's LDS readers done
    // Stage K and V chunks (64 keys x 64 dims) with async copies
#pragma unroll
    for (int i = 0; i < 8; ++i) {
      const int e   = tid + 128 * i;          // 0..1023
      const int key = e >> 4;
      const int c4  = (e & 15) * 4;
      const float* kg = qkv + ((size_t)b * TT + kc + key) * rstride
                            + CC + (size_t)h * DD + c4;
      async_copy16(lds_off(&Ks[key * SP + c4]), kg);
      async_copy16(lds_off(&Vs[key * SP + c4]), kg + CC);
    }
    if (tid < KC) {
      const int t = kc + tid;
      keepk[tid] = (t == 0 || tmask[(size_t)b * (TT - 1) + (t - 1)] != 0) ? 1.f : 0.f;
    }
    wait_async0();
    __syncthreads();

    // S = Q @ K^T  (16 x 64), with column mask
    v8f sj[4];
#pragma unroll
    for (int j = 0; j < 4; ++j) {
      v8f s = vzero;
#pragma unroll
      for (int ss = 0; ss < 16; ++ss) {
        const int d0 = 4 * ss + 2 * half;
        v2f bf;
        bf.x = Ks[(j * 16 + lm) * SP + d0];
        bf.y = Ks[(j * 16 + lm) * SP + d0 + 1];
        s = wmma_f32_4(qa[ss], bf, s);
      }
      const float km = keepk[j * 16 + lm];    // accumulator col = 16j + lane%16
#pragma unroll
      for (int r = 0; r < 8; ++r)
        s[r] = (km != 0.f) ? s[r] : -1e30f;
      sj[j] = s;
    }

    // Online softmax; rows live in 16-lane halves (wave32)
#pragma unroll
    for (int r = 0; r < 8; ++r) {
      float v = fmaxf(fmaxf(sj[0][r], sj[1][r]), fmaxf(sj[2][r], sj[3][r]));
#pragma unroll
      for (int off = 1; off < 16; off <<= 1)
        v = fmaxf(v, __shfl_xor(v, off, 32));   // stays within 16-lane half
      const float mnew = fmaxf(mrow[r], v);
      const float corr = __expf(mrow[r] - mnew);
      mrow[r] = mnew;
      lrow[r] *= corr;
#pragma unroll
      for (int j = 0; j < 4; ++j) o[j][r] *= corr;
      float ps = 0.f;
#pragma unroll
      for (int j = 0; j < 4; ++j) {
        const float p = __expf(sj[j][r] - mnew);
        sj[j][r] = p;
        ps += p;
      }
#pragma unroll
      for (int off = 1; off < 16; off <<= 1)
        ps += __shfl_xor(ps, off, 32);
      lrow[r] += ps;
    }

    // Re-layout P (accumulator layout -> A-fragment layout) through LDS
    float* pw = &Ps[wave * 16 * SP];
#pragma unroll
    for (int j = 0; j < 4; ++j)
#pragma unroll
      for (int r = 0; r < 8; ++r)
        pw[(half * 8 + r) * SP + j * 16 + lm] = sj[j][r];
    __syncthreads();

    // O += P @ V  (K dim = 64 keys)
#pragma unroll
    for (int ss = 0; ss < 16; ++ss) {
      const int k4 = 4 * ss + 2 * half;
      v2f a;
      a.x = pw[lm * SP + k4];
      a.y = pw[lm * SP + k4 + 1];
#pragma unroll
      for (int j = 0; j < 4; ++j) {
        v2f bf;
        bf.x = Vs[k4       * SP + j * 16 + lm];
        bf.y = Vs[(k4 + 1) * SP + j * 16 + lm];
        o[j] = wmma_f32_4(a, bf, o[j]);
      }
    }
  }

  // Epilogue: normalize, zero fully-masked query rows
#pragma unroll
  for (int r = 0; r < 8; ++r) {
    const int row = qt * 64 + wave * 16 + half * 8 + r;
    const bool keepq = (row == 0) || (tmask[(size_t)b * (TT - 1) + row - 1] != 0);
    const float inv = (keepq && lrow[r] > 0.f) ? (1.f / lrow[r]) : 0.f;
#pragma unroll
    for (int j = 0; j < 4; ++j) {
      const int col = h * DD + j * 16 + lm;
      y[((size_t)b * TT + row) * CC + col] = o[j][r] * inv;
    }
  }
}

// ---------------------------------------------------------------------------
extern "C" void kernel_launch(void* const* d_in, const int* in_sizes, int n_in,
                              void* d_out, int out_size, void* d_ws, size_t ws_size,
                              hipStream_t stream) {
  const float*         x     = (const float*)d_in[0];
  const unsigned char* tmask = (const unsigned char*)d_in[1];  // jnp bool -> 1 byte
  const float*         Wqkv  = (const float*)d_in[2];
  const float*         bqkv  = (const float*)d_in[3];
  const float*         Wproj = (const float*)d_in[4];
  const float*         bproj = (const float*)d_in[5];
  float* out = (float*)d_out;

  float* qkv = (float*)d_ws;                          // [B,T,3C]  96 MB
  float* yb  = qkv + (size_t)BB * TT * 3 * CC;        // [B,T,C]   32 MB

  // 1) qkv = x @ Wqkv + bqkv      (8192 x 3072, K=1024)
  dim3 g1((3 * CC) / 64, (BB * TT) / 128);
  gemm_bias_kernel<<<g1, 256, 0, stream>>>(x, Wqkv, bqkv, qkv, BB * TT, 3 * CC, CC);

  // 2) masked flash attention -> y
  dim3 g2(TT / 64, HH, BB);
  attn_kernel<<<g2, 128, 0, stream>>>(qkv, tmask, yb);

  // 3) out = y @ Wproj + bproj    (8192 x 1024, K=1024)
  dim3 g3(CC / 64, (BB * TT) / 128);
  gemm_bias_kernel<<<g3, 256, 0, stream>>>(yb, Wproj, bproj, out, BB * TT, CC, CC);
}